// CrossStockRelationship_90855738179750
// MI455X (gfx1250) — compile-verified
//
#include <hip/hip_runtime.h>

typedef float v2f __attribute__((ext_vector_type(2)));
typedef float v4f __attribute__((ext_vector_type(4)));
typedef float v8f __attribute__((ext_vector_type(8)));

#define S_STOCKS 2000
#define HID      64
#define BATCH_N  1024
#define BT       4   // batch rows per block in the reduction kernel

// ---------------------------------------------------------------------------
// Kernel A: rel_encoded = mask * (relu(rel[idx] @ W1^T + b1) @ W2^T + b2)
// One wave (32 threads) per 16 rows of S. Uses V_WMMA_F32_16X16X4_F32.
// ---------------------------------------------------------------------------
__global__ __launch_bounds__(32) void relmlp_wmma_kernel(
    const int*   __restrict__ d_idx,
    const float* __restrict__ rel,   // [S, S, H]
    const float* __restrict__ W1,    // [H, H] row-major: W1[j*H + k]
    const float* __restrict__ b1,    // [H]
    const float* __restrict__ W2,    // [H, H]
    const float* __restrict__ b2,    // [H]
    float*       __restrict__ w_out) // [S, H]
{
    __shared__ float h1[16 * HID];   // layer-1 activations tile (4 KB)

    const int idx  = d_idx[0];
    const int lane = threadIdx.x;      // 0..31
    const int m    = lane & 15;        // A-matrix row / B,C,D column within tile
    const int half = lane >> 4;        // K-split half for A/B fragments
    const int s0   = blockIdx.x * 16;  // 125 blocks * 16 = 2000 rows exactly

    const v8f zero = {0.f, 0.f, 0.f, 0.f, 0.f, 0.f, 0.f, 0.f};

    // ---------------- Layer 1: h = relu(rel_row @ W1^T + b1) ----------------
    const float* arow = rel + ((size_t)idx * S_STOCKS + (size_t)(s0 + m)) * HID;

    v8f acc[4] = {zero, zero, zero, zero};
    #pragma unroll
    for (int kk = 0; kk < 16; ++kk) {
        const int kb = kk * 4 + half * 2;
        // A fragment: lane (m, half) holds rel[s0+m][kb], rel[s0+m][kb+1]
        v2f a = *(const v2f*)(arow + kb);
        #pragma unroll
        for (int jt = 0; jt < 4; ++jt) {
            const int col = jt * 16 + m;
            // B fragment: B[k][j] = W1[j][k] (W1^T), contiguous pair in W1 row
            v2f b = *(const v2f*)(W1 + (size_t)col * HID + kb);
            acc[jt] = __builtin_amdgcn_wmma_f32_16x16x4_f32(
                false, a, false, b, (short)0, acc[jt], false, false);
        }
    }

    // bias + relu, spill to LDS so layer 2 can re-load in A-fragment layout
    #pragma unroll
    for (int jt = 0; jt < 4; ++jt) {
        const int   col  = jt * 16 + m;
        const float bias = b1[col];
        #pragma unroll
        for (int v = 0; v < 8; ++v) {
            const int r = v + half * 8;         // C/D layout: row = v + 8*half
            float x = acc[jt][v] + bias;
            h1[r * HID + col] = x > 0.f ? x : 0.f;
        }
    }
    __syncthreads();

    // ---------------- Layer 2: out = h @ W2^T + b2, then mask ---------------
    v8f acc2[4] = {zero, zero, zero, zero};
    #pragma unroll
    for (int kk = 0; kk < 16; ++kk) {
        const int kb = kk * 4 + half * 2;
        v2f a = *(const v2f*)(&h1[m * HID + kb]);
        #pragma unroll
        for (int jt = 0; jt < 4; ++jt) {
            const int col = jt * 16 + m;
            v2f b = *(const v2f*)(W2 + (size_t)col * HID + kb);
            acc2[jt] = __builtin_amdgcn_wmma_f32_16x16x4_f32(
                false, a, false, b, (short)0, acc2[jt], false, false);
        }
    }

    #pragma unroll
    for (int jt = 0; jt < 4; ++jt) {
        const int   col  = jt * 16 + m;
        const float bias = b2[col];
        #pragma unroll
        for (int v = 0; v < 8; ++v) {
            const int s = s0 + v + half * 8;
            float x = acc2[jt][v] + bias;
            if (s == idx) x = 0.f;              // exclude self row
            w_out[(size_t)s * HID + col] = x;
        }
    }
}

// ---------------------------------------------------------------------------
// Kernel B: out[b,h] = sum_s enc[b,s,h] * w[s,h]
// Pure HBM stream (524 MB of enc read once, non-temporal). w (512 KB) stays
// hot in L2 across all 1024 batch rows.
// ---------------------------------------------------------------------------
__global__ __launch_bounds__(256) void einsum_reduce_kernel(
    const float* __restrict__ enc,  // [B, S, H]
    const float* __restrict__ w,    // [S, H]
    float*       __restrict__ out)  // [B, H]
{
    const int tid = threadIdx.x;
    const int hq  = tid & 15;        // which float4 along H
    const int sg  = tid >> 4;        // s-group 0..15
    const int h4  = hq * 4;
    const int b0  = blockIdx.x * BT;

    const v4f zero4 = {0.f, 0.f, 0.f, 0.f};
    v4f acc[BT] = {zero4, zero4, zero4, zero4};

    const float* ep = enc + ((size_t)b0 * S_STOCKS + (size_t)sg) * HID + h4;
    const float* wp = w + (size_t)sg * HID + h4;

    // 2000 / 16 = 125 iterations per thread
    for (int s = sg; s < S_STOCKS; s += 16) {
        v4f wv = *(const v4f*)wp;                       // L2-resident
        #pragma unroll
        for (int u = 0; u < BT; ++u) {
            const v4f* p = (const v4f*)(ep + (size_t)u * S_STOCKS * HID);
            v4f e = __builtin_nontemporal_load(p);      // streamed once
            acc[u] += e * wv;
        }
        ep += 16 * HID;
        wp += 16 * HID;
    }

    // Cross-s-group reduction (16 partials per (u, h4)) via LDS tree
    __shared__ v4f red[256];
    #pragma unroll
    for (int u = 0; u < BT; ++u) {
        red[tid] = acc[u];
        __syncthreads();
        for (int off = 128; off >= 16; off >>= 1) {
            if (tid < off) red[tid] += red[tid + off];
            __syncthreads();
        }
        if (tid < 16) {
            *(v4f*)(out + (size_t)(b0 + u) * HID + h4) = red[tid];
        }
        __syncthreads();
    }
}

// ---------------------------------------------------------------------------
extern "C" void kernel_launch(void* const* d_in, const int* in_sizes, int n_in,
                              void* d_out, int out_size, void* d_ws, size_t ws_size,
                              hipStream_t stream) {
    (void)in_sizes; (void)n_in; (void)out_size; (void)ws_size;

    const int*   d_idx = (const int*)  d_in[0];  // stock_idx (1 element)
    const float* enc   = (const float*)d_in[1];  // [B, S, H]
    const float* rel   = (const float*)d_in[2];  // [S, S, H]
    const float* W1    = (const float*)d_in[3];  // [H, H]
    const float* b1    = (const float*)d_in[4];  // [H]
    const float* W2    = (const float*)d_in[5];  // [H, H]
    const float* b2    = (const float*)d_in[6];  // [H]
    float*       out   = (float*)d_out;          // [B, H]
    float*       w     = (float*)d_ws;           // scratch: [S, H] = 512 KB

    relmlp_wmma_kernel<<<S_STOCKS / 16, 32, 0, stream>>>(
        d_idx, rel, W1, b1, W2, b2, w);

    einsum_reduce_kernel<<<BATCH_N / BT, 256, 0, stream>>>(enc, w, out);
}